// S2S_4234837754574
// MI455X (gfx1250) — compile-verified
//
#include <hip/hip_runtime.h>
#include <math.h>

// Problem constants (match reference)
#define B_SZ   64
#define T_LEN  80
#define D_IN   4096
#define H_SZ   256
#define G4     1024      // 4*H
#define V_SZ   6144
#define L_DEC  30
#define SOS    1

typedef float v2f __attribute__((ext_vector_type(2)));
typedef float v8f __attribute__((ext_vector_type(8)));

// CDNA5 fp32 WMMA: D(16x16 f32) = A(16x4 f32) * B(4x16 f32) + C
__device__ __forceinline__ v8f wmma_f32(v2f a, v2f b, v8f c) {
  return __builtin_amdgcn_wmma_f32_16x16x4_f32(
      /*neg_a=*/false, a, /*neg_b=*/false, b,
      /*c_mod=*/(short)0, c, /*reuse_a=*/false, /*reuse_b=*/false);
}

__device__ __forceinline__ float sigmoidf_(float x) {
  return 1.0f / (1.0f + expf(-x));
}

// -------- gfx1250 async global->LDS staging (ASYNCcnt path), with fallback ---
#if defined(__gfx1250__)
#if __has_builtin(__builtin_amdgcn_global_load_async_to_lds_b128) && \
    __has_builtin(__builtin_amdgcn_s_wait_asynccnt)
#define HAVE_ASYNC_LDS 1
#else
#define HAVE_ASYNC_LDS 0
#endif
#else
#define HAVE_ASYNC_LDS 0
#endif

#if HAVE_ASYNC_LDS
typedef int b128_t __attribute__((vector_size(16)));
typedef __attribute__((address_space(1))) b128_t* gptr_b128;   // global int4*
typedef __attribute__((address_space(3))) b128_t* lptr_b128;   // LDS int4*
#endif

__device__ __forceinline__ void async_cp16(void* lds_dst, const void* gsrc) {
#if HAVE_ASYNC_LDS
  // Low 32 bits of a flat LDS pointer are the LDS-aperture offset.
  __builtin_amdgcn_global_load_async_to_lds_b128(
      (gptr_b128)(unsigned long long)(uintptr_t)gsrc,
      (lptr_b128)(unsigned)(uintptr_t)lds_dst,
      0, 0);
#else
  *(float4*)lds_dst = *(const float4*)gsrc;
#endif
}

__device__ __forceinline__ void async_wait_all() {
#if HAVE_ASYNC_LDS
  __builtin_amdgcn_s_wait_asynccnt(0);
#endif
}

// ---------------------------------------------------------------------------
// Phase 1: OUT[m][n] = X[m][:] . W[n][:] + (bih[n] + bhh[n])
//   X: [5120][4096]  W: [1024][4096]  OUT: [5120][1024]
// Block = 256 thr (8 waves), macro-tile 64x128, Kc=32.
// Waves in 2x4 grid; each wave owns a 32x32 register tile (4 accumulators):
// 4 WMMAs per 4 fragment loads -> half the LDS traffic per WMMA.
// Both As and Bs are stored [row][k] (global-contiguous along k, pad to 36
// floats = 144 B so every 16 B chunk is aligned; 36m mod 64 banks is
// conflict-free for m in [0,16)).
// ---------------------------------------------------------------------------
__global__ __launch_bounds__(256) void k_xproj(
    const float* __restrict__ X, const float* __restrict__ W,
    const float* __restrict__ bih, const float* __restrict__ bhh,
    float* __restrict__ OUT)
{
  const int K = D_IN;
  const int KC = 32;
  __shared__ float As[64][36];     // As[m][k]
  __shared__ float Bs[128][36];    // Bs[n][k]

  const int tid  = threadIdx.x;
  const int wave = tid >> 5, lane = tid & 31;
  const int bm = blockIdx.x * 64;     // 80 blocks over M=5120
  const int bn = blockIdx.y * 128;    // 8  blocks over N=1024
  const int mw = (wave >> 2) << 5;    // 0 | 32
  const int nw = (wave & 3) << 5;     // 0,32,64,96
  const int lm = lane & 15, kb = (lane >> 4) << 1;

  v8f acc00 = {0.f,0.f,0.f,0.f,0.f,0.f,0.f,0.f};
  v8f acc01 = acc00, acc10 = acc00, acc11 = acc00;

  for (int k0 = 0; k0 < K; k0 += KC) {
    // Stage A: 64x32 floats = 512 x 16B chunks, 2 per thread (async)
    #pragma unroll
    for (int q = 0; q < 2; ++q) {
      const int ch = tid + (q << 8);
      const int r = ch >> 3, c = (ch & 7) << 2;
      async_cp16(&As[r][c], X + (size_t)(bm + r) * K + k0 + c);
    }
    // Stage B: 128x32 floats = 1024 x 16B chunks, 4 per thread (async)
    #pragma unroll
    for (int q = 0; q < 4; ++q) {
      const int ch = tid + (q << 8);
      const int r = ch >> 3, c = (ch & 7) << 2;
      async_cp16(&Bs[r][c], W + (size_t)(bn + r) * K + k0 + c);
    }
    async_wait_all();
    __syncthreads();

    #pragma unroll
    for (int kk = 0; kk < KC; kk += 4) {
      v2f a0, a1, b0, b1;
      a0.x = As[mw + lm][kk + kb];      a0.y = As[mw + lm][kk + kb + 1];
      a1.x = As[mw + 16 + lm][kk + kb]; a1.y = As[mw + 16 + lm][kk + kb + 1];
      b0.x = Bs[nw + lm][kk + kb];      b0.y = Bs[nw + lm][kk + kb + 1];
      b1.x = Bs[nw + 16 + lm][kk + kb]; b1.y = Bs[nw + 16 + lm][kk + kb + 1];
      acc00 = wmma_f32(a0, b0, acc00);
      acc01 = wmma_f32(a0, b1, acc01);
      acc10 = wmma_f32(a1, b0, acc10);
      acc11 = wmma_f32(a1, b1, acc11);
    }
    __syncthreads();
  }

  // Epilogue: bias + store 4 tiles
  const int eoff = (lane >> 4) << 3;
  #pragma unroll
  for (int tm = 0; tm < 2; ++tm) {
    #pragma unroll
    for (int tn = 0; tn < 2; ++tn) {
      const v8f acc = (tm == 0) ? (tn == 0 ? acc00 : acc01)
                                : (tn == 0 ? acc10 : acc11);
      const int col = bn + nw + (tn << 4) + lm;
      const float bb = bih[col] + bhh[col];
      const int rb = bm + mw + (tm << 4) + eoff;
      #pragma unroll
      for (int e = 0; e < 8; ++e)
        OUT[(size_t)(rb + e) * G4 + col] = acc[e] + bb;
    }
  }
}

// ---------------------------------------------------------------------------
// Phase 2: encoder recurrence. Single persistent workgroup, 1024 thr.
// Each wave owns 2 "gate groups": (m-tile, j-tile) with all 4 gate columns,
// so c-state lives in registers and h is double-buffered in LDS.
// ---------------------------------------------------------------------------
__global__ __launch_bounds__(1024) void k_encoder(
    const float* __restrict__ XP,    // [B*T][4H], biases folded in
    const float* __restrict__ Whh,   // [4H][H]
    float* __restrict__ h_out, float* __restrict__ c_out)   // [B][H] each
{
  extern __shared__ float lds[];
  float* hbuf = lds;                               // [2][B][H] = 128 KB
  const int tid  = threadIdx.x;
  const int wave = tid >> 5, lane = tid & 31;
  const int lm = lane & 15, kb = (lane >> 4) << 1, eoff = (lane >> 4) << 3;

  for (int i = tid; i < B_SZ * H_SZ; i += 1024) hbuf[i] = 0.0f;
  v8f cfrag[2];
  #pragma unroll
  for (int gi = 0; gi < 2; ++gi)
    #pragma unroll
    for (int e = 0; e < 8; ++e) cfrag[gi][e] = 0.0f;
  __syncthreads();

  for (int t = 0; t < T_LEN; ++t) {
    const float* hcur = hbuf + (t & 1) * (B_SZ * H_SZ);
    float* hnext      = hbuf + ((t + 1) & 1) * (B_SZ * H_SZ);
    #pragma unroll
    for (int gi = 0; gi < 2; ++gi) {
      const int g  = wave + 32 * gi;
      const int m0 = (g >> 4) << 4;
      const int jc = ((g & 15) << 4) + lm;
      v8f ai, af, ag, ao;
      #pragma unroll
      for (int e = 0; e < 8; ++e) {
        const float* xr = XP + ((size_t)(m0 + eoff + e) * T_LEN + t) * G4;
        ai[e] = xr[jc];
        af[e] = xr[H_SZ + jc];
        ag[e] = xr[2 * H_SZ + jc];
        ao[e] = xr[3 * H_SZ + jc];
        if (t + 1 < T_LEN) __builtin_prefetch(xr + G4, 0, 0);
      }
      const float* hr = hcur + (m0 + lm) * H_SZ;
      const float* wi = Whh + (size_t)(jc) * H_SZ;
      const float* wf = Whh + (size_t)(H_SZ + jc) * H_SZ;
      const float* wg = Whh + (size_t)(2 * H_SZ + jc) * H_SZ;
      const float* wo = Whh + (size_t)(3 * H_SZ + jc) * H_SZ;
      for (int k = 0; k < H_SZ; k += 4) {
        v2f a, b;
        a.x = hr[k + kb]; a.y = hr[k + kb + 1];
        b.x = wi[k + kb]; b.y = wi[k + kb + 1]; ai = wmma_f32(a, b, ai);
        b.x = wf[k + kb]; b.y = wf[k + kb + 1]; af = wmma_f32(a, b, af);
        b.x = wg[k + kb]; b.y = wg[k + kb + 1]; ag = wmma_f32(a, b, ag);
        b.x = wo[k + kb]; b.y = wo[k + kb + 1]; ao = wmma_f32(a, b, ao);
      }
      #pragma unroll
      for (int e = 0; e < 8; ++e) {
        const float iv = sigmoidf_(ai[e]);
        const float fv = sigmoidf_(af[e]);
        const float gv = tanhf(ag[e]);
        const float ov = sigmoidf_(ao[e]);
        const float cv = fv * cfrag[gi][e] + iv * gv;
        cfrag[gi][e] = cv;
        hnext[(m0 + eoff + e) * H_SZ + jc] = ov * tanhf(cv);
      }
    }
    __syncthreads();
  }

  const float* hf = hbuf + (T_LEN & 1) * (B_SZ * H_SZ);
  for (int i = tid; i < B_SZ * H_SZ; i += 1024) h_out[i] = hf[i];
  #pragma unroll
  for (int gi = 0; gi < 2; ++gi) {
    const int g  = wave + 32 * gi;
    const int m0 = (g >> 4) << 4, j0 = (g & 15) << 4;
    #pragma unroll
    for (int e = 0; e < 8; ++e)
      c_out[(m0 + eoff + e) * H_SZ + j0 + lm] = cfrag[gi][e];
  }
}

// ---------------------------------------------------------------------------
// Decoder helpers
// ---------------------------------------------------------------------------
__device__ __forceinline__ void logits_pass(
    const float* __restrict__ A,       // LDS [64][256] (h2 or embed rows)
    float* __restrict__ chunk,         // LDS [64][512]
    float* __restrict__ bestv, int* __restrict__ besti,
    const float* __restrict__ Wout, const float* __restrict__ bout,
    float* __restrict__ out_logits, int emit,
    int tid, int wave, int lane)
{
  const int lm = lane & 15, kb = (lane >> 4) << 1, eoff = (lane >> 4) << 3;
  if (tid < B_SZ) { bestv[tid] = -INFINITY; besti[tid] = 0x7FFFFFFF; }
  __syncthreads();

  for (int c0 = 0; c0 < V_SZ; c0 += 512) {
    #pragma unroll
    for (int ti = 0; ti < 4; ++ti) {
      const int tile = ti * 32 + wave;          // 128 tiles / 32 waves
      const int m0 = (tile >> 5) << 4;
      const int n0 = (tile & 31) << 4;
      const int col = c0 + n0 + lm;
      const float bb = bout[col];
      v8f acc;
      #pragma unroll
      for (int e = 0; e < 8; ++e) acc[e] = bb;
      const float* ar = A + (m0 + lm) * H_SZ;
      const float* wr = Wout + (size_t)col * H_SZ;
      for (int k = 0; k < H_SZ; k += 4) {
        v2f a, b;
        a.x = ar[k + kb]; a.y = ar[k + kb + 1];
        b.x = wr[k + kb]; b.y = wr[k + kb + 1];
        acc = wmma_f32(a, b, acc);
      }
      #pragma unroll
      for (int e = 0; e < 8; ++e) {
        const int row = m0 + eoff + e;
        chunk[row * 512 + n0 + lm] = acc[e];
        out_logits[((size_t)row * L_DEC + emit) * V_SZ + col] = acc[e];
      }
    }
    __syncthreads();

    // Partial argmax over this 512-col chunk: 16 lanes per row, 2 rows/wave
    const int r = wave * 2 + (lane >> 4);
    float bv = -INFINITY; int bi = 0x7FFFFFFF;
    for (int c = (lane & 15); c < 512; c += 16) {
      const float v = chunk[r * 512 + c];
      if (v > bv) { bv = v; bi = c0 + c; }       // ascending c => first max kept
    }
    #pragma unroll
    for (int m = 8; m >= 1; m >>= 1) {
      const float ov = __shfl_xor(bv, m, 32);
      const int   oi = __shfl_xor(bi, m, 32);
      if (ov > bv || (ov == bv && oi < bi)) { bv = ov; bi = oi; }
    }
    if ((lane & 15) == 0) {
      if (bv > bestv[r] || (bv == bestv[r] && bi < besti[r])) {
        bestv[r] = bv; besti[r] = bi;
      }
    }
    __syncthreads();
  }
}

__device__ __forceinline__ void lstm2_step(
    float* __restrict__ h2, const float* __restrict__ xb,
    const float* __restrict__ Wih, const float* __restrict__ Whh,
    const float* __restrict__ bih, const float* __restrict__ bhh,
    v8f (&cfrag)[2], int wave, int lane)
{
  const int lm = lane & 15, kb = (lane >> 4) << 1, eoff = (lane >> 4) << 3;
  v8f hv[2];
  #pragma unroll
  for (int gi = 0; gi < 2; ++gi) {
    const int g  = wave + 32 * gi;
    const int m0 = (g >> 4) << 4;
    const int jc = ((g & 15) << 4) + lm;
    v8f ai, af, ag, ao;
    const float b_i = bih[jc] + bhh[jc];
    const float b_f = bih[H_SZ + jc] + bhh[H_SZ + jc];
    const float b_g = bih[2 * H_SZ + jc] + bhh[2 * H_SZ + jc];
    const float b_o = bih[3 * H_SZ + jc] + bhh[3 * H_SZ + jc];
    #pragma unroll
    for (int e = 0; e < 8; ++e) { ai[e] = b_i; af[e] = b_f; ag[e] = b_g; ao[e] = b_o; }

    const float* xr  = xb + (m0 + lm) * H_SZ;
    const float* hr  = h2 + (m0 + lm) * H_SZ;
    const float* wii = Wih + (size_t)(jc) * H_SZ;
    const float* wif = Wih + (size_t)(H_SZ + jc) * H_SZ;
    const float* wig = Wih + (size_t)(2 * H_SZ + jc) * H_SZ;
    const float* wio = Wih + (size_t)(3 * H_SZ + jc) * H_SZ;
    const float* whi = Whh + (size_t)(jc) * H_SZ;
    const float* whf = Whh + (size_t)(H_SZ + jc) * H_SZ;
    const float* whg = Whh + (size_t)(2 * H_SZ + jc) * H_SZ;
    const float* who = Whh + (size_t)(3 * H_SZ + jc) * H_SZ;

    for (int k = 0; k < H_SZ; k += 4) {
      v2f a, b, ah;
      a.x  = xr[k + kb]; a.y  = xr[k + kb + 1];
      ah.x = hr[k + kb]; ah.y = hr[k + kb + 1];
      b.x = wii[k + kb]; b.y = wii[k + kb + 1]; ai = wmma_f32(a,  b, ai);
      b.x = whi[k + kb]; b.y = whi[k + kb + 1]; ai = wmma_f32(ah, b, ai);
      b.x = wif[k + kb]; b.y = wif[k + kb + 1]; af = wmma_f32(a,  b, af);
      b.x = whf[k + kb]; b.y = whf[k + kb + 1]; af = wmma_f32(ah, b, af);
      b.x = wig[k + kb]; b.y = wig[k + kb + 1]; ag = wmma_f32(a,  b, ag);
      b.x = whg[k + kb]; b.y = whg[k + kb + 1]; ag = wmma_f32(ah, b, ag);
      b.x = wio[k + kb]; b.y = wio[k + kb + 1]; ao = wmma_f32(a,  b, ao);
      b.x = who[k + kb]; b.y = who[k + kb + 1]; ao = wmma_f32(ah, b, ao);
    }
    #pragma unroll
    for (int e = 0; e < 8; ++e) {
      const float iv = sigmoidf_(ai[e]);
      const float fv = sigmoidf_(af[e]);
      const float gv = tanhf(ag[e]);
      const float ov = sigmoidf_(ao[e]);
      const float cv = fv * cfrag[gi][e] + iv * gv;
      cfrag[gi][e] = cv;
      hv[gi][e] = ov * tanhf(cv);
    }
  }
  __syncthreads();                 // all reads of old h2 complete
  #pragma unroll
  for (int gi = 0; gi < 2; ++gi) {
    const int g  = wave + 32 * gi;
    const int m0 = (g >> 4) << 4, j0 = (g & 15) << 4;
    #pragma unroll
    for (int e = 0; e < 8; ++e)
      h2[(m0 + eoff + e) * H_SZ + j0 + lm] = hv[gi][e];
  }
  __syncthreads();
}

// ---------------------------------------------------------------------------
// Phase 3: greedy autoregressive decoder. Single persistent workgroup.
// ---------------------------------------------------------------------------
__global__ __launch_bounds__(1024) void k_decoder(
    const float* __restrict__ h_enc, const float* __restrict__ c_enc,
    const float* __restrict__ embed,
    const float* __restrict__ Wout, const float* __restrict__ bout,
    const float* __restrict__ Wih2, const float* __restrict__ Whh2,
    const float* __restrict__ bih2, const float* __restrict__ bhh2,
    float* __restrict__ out_logits,  // [64][30][6144]
    float* __restrict__ out_sym)     // [64][30] (stored as float)
{
  extern __shared__ float lds[];
  float* h2    = lds;                      // [64][256]
  float* xb    = lds + B_SZ * H_SZ;        // [64][256]
  float* chunk = lds + 2 * B_SZ * H_SZ;    // [64][512]
  float* bestv = chunk + B_SZ * 512;       // [64]
  int*   besti = (int*)(bestv + B_SZ);     // [64]
  int*   symb  = besti + B_SZ;             // [64]

  const int tid  = threadIdx.x;
  const int wave = tid >> 5, lane = tid & 31;
  const int lm = lane & 15, eoff = (lane >> 4) << 3;

  for (int i = tid; i < B_SZ * H_SZ; i += 1024) {
    h2[i] = h_enc[i];
    xb[i] = embed[SOS * H_SZ + (i & (H_SZ - 1))];
  }
  v8f cfrag[2];
  #pragma unroll
  for (int gi = 0; gi < 2; ++gi) {
    const int g  = wave + 32 * gi;
    const int m0 = (g >> 4) << 4, j0 = (g & 15) << 4;
    #pragma unroll
    for (int e = 0; e < 8; ++e)
      cfrag[gi][e] = c_enc[(m0 + eoff + e) * H_SZ + j0 + lm];
  }
  if (tid < B_SZ) symb[tid] = SOS;
  __syncthreads();

  // Emission 0: dec_o0 = embed[SOS] @ Wout.T + bout ; symbol 0 = SOS
  logits_pass(xb, chunk, bestv, besti, Wout, bout, out_logits, 0, tid, wave, lane);
  if (tid < B_SZ) out_sym[tid * L_DEC + 0] = (float)SOS;

  for (int l = 0; l < L_DEC - 1; ++l) {
    __syncthreads();
    // x = embed[sym] : 16 threads per row, 16 floats each
    {
      const int m = tid >> 4, c = (tid & 15) << 4;
      const float* er = embed + (size_t)symb[m] * H_SZ + c;
      float* xw = xb + m * H_SZ + c;
      #pragma unroll
      for (int q = 0; q < 16; q += 4)
        *(float4*)(xw + q) = *(const float4*)(er + q);
    }
    __syncthreads();
    lstm2_step(h2, xb, Wih2, Whh2, bih2, bhh2, cfrag, wave, lane);
    logits_pass(h2, chunk, bestv, besti, Wout, bout, out_logits, l + 1, tid, wave, lane);
    if (tid < B_SZ) {
      symb[tid] = besti[tid];
      out_sym[tid * L_DEC + (l + 1)] = (float)besti[tid];
    }
  }
}

// ---------------------------------------------------------------------------
extern "C" void kernel_launch(void* const* d_in, const int* in_sizes, int n_in,
                              void* d_out, int out_size, void* d_ws, size_t ws_size,
                              hipStream_t stream) {
  const float* input = (const float*)d_in[0];   // [64][80][4096]
  // d_in[1] = target_lengths (all 30, compile-time constant here)
  const float* embed = (const float*)d_in[2];   // [6144][256]
  const float* W_out = (const float*)d_in[3];   // [6144][256]
  const float* b_out = (const float*)d_in[4];   // [6144]
  const float* W_ih1 = (const float*)d_in[5];   // [1024][4096]
  const float* W_hh1 = (const float*)d_in[6];   // [1024][256]
  const float* b_ih1 = (const float*)d_in[7];
  const float* b_hh1 = (const float*)d_in[8];
  const float* W_ih2 = (const float*)d_in[9];   // [1024][256]
  const float* W_hh2 = (const float*)d_in[10];  // [1024][256]
  const float* b_ih2 = (const float*)d_in[11];
  const float* b_hh2 = (const float*)d_in[12];

  float* out   = (float*)d_out;
  float* ws    = (float*)d_ws;
  float* XP    = ws;                               // [5120][1024]
  float* h_enc = ws + (size_t)B_SZ * T_LEN * G4;   // [64][256]
  float* c_enc = h_enc + B_SZ * H_SZ;              // [64][256]

  // Phase 1: big parallel GEMM (input projection, biases folded)
  dim3 g1((B_SZ * T_LEN) / 64, G4 / 128);
  k_xproj<<<g1, dim3(256), 0, stream>>>(input, W_ih1, b_ih1, b_hh1, XP);

  // Phase 2: sequential encoder recurrence (persistent WG, 128KB dyn LDS)
  k_encoder<<<dim3(1), dim3(1024), 2 * B_SZ * H_SZ * sizeof(float), stream>>>(
      XP, W_hh1, h_enc, c_enc);

  // Phase 3: autoregressive decoder (persistent WG, ~257KB dyn LDS)
  const size_t dec_lds =
      (size_t)(2 * B_SZ * H_SZ + B_SZ * 512 + B_SZ) * sizeof(float) +
      (size_t)(2 * B_SZ) * sizeof(int);
  k_decoder<<<dim3(1), dim3(1024), dec_lds, stream>>>(
      h_enc, c_enc, embed, W_out, b_out, W_ih2, W_hh2, b_ih2, b_hh2,
      out, out + (size_t)B_SZ * L_DEC * V_SZ);
}